// ConvLogicTreeLayer_81123342287623
// MI455X (gfx1250) — compile-verified
//
#include <hip/hip_runtime.h>

// ---------------------------------------------------------------------------
// ConvLogicTreeLayer for gfx1250 (MI455X).
//
// Math reduction: each tree node = sum_g softmax(w)_g * op_g(a,b), and every
// op_g is affine in {1, a, b, ab}.  So node(a,b) = C0 + C1*a + C2*b + C3*ab
// with C[] precomputed per (out_channel, gate).  Per output element: 7 nodes
// x 4 FMA = trivial compute; the problem is pure data movement + gather.
//
// CDNA5 path: per workgroup, one 3-D TENSOR_LOAD_TO_LDS stages the haloed
// input tile (rows+2) x 64ch x 32col into LDS.  D# pad_enable inserts 2
// padding dwords after every 32 dwords -> LDS row stride 34; with halo cells
// pre-zeroed, the conv's zero padding is free.  Completion via
// s_wait_tensorcnt.  Leaf gathers are ds_load_b32; per-channel constants are
// wave-uniform scalar loads.
// ---------------------------------------------------------------------------

#define TH 4                         // output rows per workgroup
#define LDS_ROWS (TH + 2)            // with top/bottom halo
#define LDS_ROW_STRIDE (64 * 34)     // [row][channel][34] layout, 2176 floats
#define LDS_ELEMS (LDS_ROWS * LDS_ROW_STRIDE)  // 13056 floats = 52,224 B

typedef __attribute__((ext_vector_type(4))) unsigned int v4u;
typedef __attribute__((ext_vector_type(8))) int          v8i;
typedef __attribute__((ext_vector_type(4))) int          v4i;

// ---------------------------------------------------------------------------
// Prep kernel: softmax(weights[o,g,:16]) folded into 4 affine coefficients,
// and leaf_indices decoded into LDS gather offsets.
//   off2[o][j] = ki*LDS_ROW_STRIDE + c*34 + kj   (leaf = tile[base + off2])
// ---------------------------------------------------------------------------
__global__ __launch_bounds__(256) void logic_prep_kernel(
    const float* __restrict__ W,      // [128][7][16]
    const int*   __restrict__ leaf,   // [128][8]
    float*       __restrict__ coef,   // [128][7][4]
    int*         __restrict__ off2)   // [128][8]
{
    const int tid = blockIdx.x * blockDim.x + threadIdx.x;

    if (tid < 128 * 7) {
        const float* w = W + tid * 16;
        float mx = w[0];
        #pragma unroll
        for (int i = 1; i < 16; ++i) mx = fmaxf(mx, w[i]);
        float e[16];
        float s = 0.0f;
        #pragma unroll
        for (int i = 0; i < 16; ++i) { e[i] = __expf(w[i] - mx); s += e[i]; }
        const float inv = 1.0f / s;
        // op_g = T0 + T1*a + T2*b + T3*ab for the 16 reference binary ops
        const float T1[16] = {0,0,1,1,0,0,1,1,-1,-1,0,0,-1,-1,0,0};
        const float T2[16] = {0,0,0,0,1,1,1,1,-1,-1,-1,-1,0,0,0,0};
        const float T3[16] = {0,1,-1,0,-1,0,-2,-1,1,2,0,1,0,1,-1,0};
        float c0 = 0.f, c1 = 0.f, c2 = 0.f, c3 = 0.f;
        #pragma unroll
        for (int i = 0; i < 16; ++i) {
            const float wi = e[i] * inv;
            if (i >= 8) c0 += wi;             // T0 = 1 for ops 8..15
            c1 += wi * T1[i];
            c2 += wi * T2[i];
            c3 += wi * T3[i];
        }
        coef[tid * 4 + 0] = c0;
        coef[tid * 4 + 1] = c1;
        coef[tid * 4 + 2] = c2;
        coef[tid * 4 + 3] = c3;
    }

    if (tid < 128 * 8) {
        const int idx = leaf[tid];            // c*9 + ki*3 + kj, in [0,576)
        const int c  = idx / 9;
        const int kp = idx % 9;
        const int ki = kp / 3;
        const int kj = kp % 3;
        off2[tid] = ki * LDS_ROW_STRIDE + c * 34 + kj;
    }
}

__device__ __forceinline__ float logic_gate(float a, float b, const float* c) {
    return fmaf(c[3], a * b, fmaf(c[2], b, fmaf(c[1], a, c[0])));
}

// ---------------------------------------------------------------------------
// Main kernel.  grid = (32/TH, 16) = (8, 16); block = 256 (8 wave32).
// ---------------------------------------------------------------------------
__global__ __launch_bounds__(256) void logic_tree_main_kernel(
    const float* __restrict__ x,      // [16][64][32][32]
    const float* __restrict__ coef,   // [128][7][4]
    const int*   __restrict__ off2,   // [128][8]
    float*       __restrict__ out)    // [16][128][32][32]
{
    __shared__ float tile[LDS_ELEMS];

    const int r0 = blockIdx.x * TH;   // first output row of this tile
    const int b  = blockIdx.y;
    const int t  = threadIdx.x;

    // Zero ONLY the halo cells (TDM overwrites every interior data column):
    //  - columns 0 and 33 of each (row, channel) pane (left/right conv pad)
    //  - the whole top/bottom halo row-block when it falls outside the image
    for (int i = t; i < LDS_ROWS * 64; i += 256) {
        const int p = i * 34;
        tile[p]      = 0.0f;
        tile[p + 33] = 0.0f;
    }
    if (r0 == 0) {
        for (int i = t; i < LDS_ROW_STRIDE; i += 256) tile[i] = 0.0f;
    }
    if (r0 == 32 - TH) {
        for (int i = t; i < LDS_ROW_STRIDE; i += 256)
            tile[(LDS_ROWS - 1) * LDS_ROW_STRIDE + i] = 0.0f;
    }
    __syncthreads();

    if (t == 0) {
        // Valid source rows (global image rows), clamped at the borders.
        int rstart = r0 - 1; if (rstart < 0) rstart = 0;
        int rend   = r0 + TH; if (rend > 31) rend = 31;
        const int rows     = rend - rstart + 1;
        const int lds_row0 = rstart - (r0 - 1);      // 1 iff top tile

        // Low 32 bits of a flat pointer to LDS == byte offset in LDS.
        const unsigned lds_base = (unsigned)(unsigned long long)(void*)tile;
        const unsigned lds_addr =
            lds_base + (unsigned)((lds_row0 * LDS_ROW_STRIDE + 1) * 4);

        const unsigned long long gaddr = (unsigned long long)(const void*)
            (x + ((size_t)b * 64u * 1024u + (size_t)rstart * 32u));

        // ---- Tensor DMA descriptor (D#) ----
        // 3-D: dim0 = 32 cols (fastest), dim1 = 64 channels (stride 1024),
        //      dim2 = rows (stride 32).  data_size = 4B.
        // pad_enable: after every 32 dwords written, skip 2 dwords ->
        // per-(row,channel) LDS stride 34, data at cols 1..32.
        v4u g0;
        g0.x = 1u;                                      // count=1 (valid)
        g0.y = lds_addr;                                // lds_addr (bytes)
        g0.z = (unsigned)(gaddr & 0xFFFFFFFFu);         // global_addr[31:0]
        g0.w = (unsigned)((gaddr >> 32) & 0x01FFFFFFu)  // global_addr[56:32]
             | (2u << 30);                              // type = 2 ("image")

        v8i g1;
        g1[0] = (int)((2u << 16)        // data_size = 4B
                    | (1u << 20)        // pad_enable
                    | (4u << 22)        // pad_interval: 2^(4+1)=32 dwords
                    | (1u << 25));      // pad_amount: 2 dwords
        g1[1] = (int)(32u << 16);       // tensor_dim0 = 32 (W)
        g1[2] = (int)(64u << 16);       // tensor_dim1 = 64 (C)
        g1[3] = (int)(32u << 16);       // tile_dim0   = 32 cols
        g1[4] = (int)(64u | ((unsigned)rows << 16));  // tile_dim1=64 ch, tile_dim2=rows
        g1[5] = 1024;                   // tensor_dim0_stride = channel stride
        g1[6] = (int)(32u << 16);       // tensor_dim1_stride = row stride (32)
        g1[7] = 0;

        v4i g2; g2.x = 32; g2.y = 0; g2.z = 0; g2.w = 0;  // tensor_dim2 = 32
        v4i g3; g3.x = 0;  g3.y = 0;  g3.z = 0;  g3.w = 0;

#if defined(__clang_major__) && (__clang_major__ >= 23)
        v8i gz = {0, 0, 0, 0, 0, 0, 0, 0};
        __builtin_amdgcn_tensor_load_to_lds(g0, g1, g2, g3, gz, 0);
#else
        __builtin_amdgcn_tensor_load_to_lds(g0, g1, g2, g3, 0);
#endif
        __builtin_amdgcn_s_wait_tensorcnt(0);
    }
    __syncthreads();

    // Threads 0..127 -> positions, o in [0,64); 128..255 -> o in [64,128).
    // o is wave-uniform (waves are 32-thread aligned within each half).
    const int pos   = t & 127;
    const int hl    = pos >> 5;               // local row 0..3
    const int w     = pos & 31;               // col 0..31
    const int obase = (t >> 7) << 6;          // 0 or 64
    const int lbase = hl * LDS_ROW_STRIDE + w;

    float* outp = out + (((size_t)b * 128u + (unsigned)obase) * 32u
                         + (unsigned)(r0 + hl)) * 32u + (unsigned)w;

    #pragma unroll 2
    for (int i = 0; i < 64; ++i) {
        const int o = __builtin_amdgcn_readfirstlane(obase + i);  // scalarize
        const int*   op = off2 + o * 8;       // -> s_load
        const float* cf = coef + o * 28;      // -> s_load (gates 0..3 used)

        const float l0 = tile[lbase + op[0]];
        const float l1 = tile[lbase + op[1]];
        const float l2 = tile[lbase + op[2]];
        const float l3 = tile[lbase + op[3]];
        const float l4 = tile[lbase + op[4]];
        const float l5 = tile[lbase + op[5]];
        const float l6 = tile[lbase + op[6]];
        const float l7 = tile[lbase + op[7]];

        // level 0: gates 0..3  (reference gate_ids = 2^level-1 + arange)
        const float n0 = logic_gate(l0, l1, cf + 0);
        const float n1 = logic_gate(l2, l3, cf + 4);
        const float n2 = logic_gate(l4, l5, cf + 8);
        const float n3 = logic_gate(l6, l7, cf + 12);
        // level 1: gates 1,2
        const float m0 = logic_gate(n0, n1, cf + 4);
        const float m1 = logic_gate(n2, n3, cf + 8);
        // level 2: gate 3
        const float r  = logic_gate(m0, m1, cf + 12);

        outp[(size_t)i * 1024u] = r;
    }
}

// ---------------------------------------------------------------------------
extern "C" void kernel_launch(void* const* d_in, const int* in_sizes, int n_in,
                              void* d_out, int out_size, void* d_ws, size_t ws_size,
                              hipStream_t stream) {
    const float* x    = (const float*)d_in[0];   // (16,64,32,32) f32
    const float* W    = (const float*)d_in[1];   // (128,7,16) f32
    const int*   leaf = (const int*)d_in[2];     // (128,8) i32
    float*       out  = (float*)d_out;           // (16,128,32,32) f32

    float* coef = (float*)d_ws;                              // 128*7*4 f32
    int*   off2 = (int*)((char*)d_ws + 128 * 7 * 4 * sizeof(float));

    logic_prep_kernel<<<4, 256, 0, stream>>>(W, leaf, coef, off2);

    dim3 grid(32 / TH, 16);                      // 8 row-tiles x 16 batches
    logic_tree_main_kernel<<<grid, 256, 0, stream>>>(x, coef, off2, out);
}